// VectorQuantizer_41729902248341
// MI455X (gfx1250) — compile-verified
//
#include <hip/hip_runtime.h>

typedef float v2f __attribute__((ext_vector_type(2)));
typedef float v8f __attribute__((ext_vector_type(8)));

#define LATENT_DIM   64
#define NUM_EMB      1024
#define WAVES_PER_BLOCK 8
#define ROWS_PER_WAVE   32      // two 16-row M-tiles per wave
#define ROWS_PER_BLOCK  256
#define BLOCK_THREADS   256

// LDS layout in floats: codebook | e^2 | per-block row indices
#define ESQ_OFF   (NUM_EMB * LATENT_DIM)         // 65536
#define IDX_OFF   (ESQ_OFF + NUM_EMB)            // 66560
#define LDS_FLOATS (IDX_OFF + ROWS_PER_BLOCK)    // 66816 floats = 267264 B (< 320 KB)

extern "C" __global__ __launch_bounds__(BLOCK_THREADS, 1)
void vq_kernel(const float* __restrict__ x,      // [N, 64]
               const float* __restrict__ cb,     // [1024, 64]
               int*   __restrict__ out_idx,      // [N]
               float* __restrict__ out_q) {      // [N, 64]
  extern __shared__ float lds[];
  float* cbL  = lds;                 // 1024 x 64
  float* esq  = lds + ESQ_OFF;       // 1024
  int*   idxb = (int*)(lds + IDX_OFF);

  const int tid = threadIdx.x;

  // ---- stage codebook into LDS (256 KB), float4 per thread ----
  {
    const float4* src = (const float4*)cb;
    float4* dst = (float4*)cbL;
    #pragma unroll 4
    for (int i = tid; i < NUM_EMB * LATENT_DIM / 4; i += BLOCK_THREADS)
      dst[i] = src[i];
  }
  // ---- ||e||^2 per code (read from global; overlaps staging) ----
  for (int c = tid; c < NUM_EMB; c += BLOCK_THREADS) {
    const float4* row = (const float4*)(cb + c * LATENT_DIM);
    float s = 0.f;
    #pragma unroll
    for (int i = 0; i < LATENT_DIM / 4; ++i) {
      float4 v = row[i];
      s = fmaf(v.x, v.x, s); s = fmaf(v.y, v.y, s);
      s = fmaf(v.z, v.z, s); s = fmaf(v.w, v.w, s);
    }
    esq[c] = s;
  }
  __syncthreads();

  const int lane = tid & 31;
  const int wave = tid >> 5;
  const int n = lane & 15;   // column (code) within tile / row within M-tile
  const int h = lane >> 4;   // lane-half: selects K parity pair (K = 2h + v)
  const long rowBase = (long)blockIdx.x * ROWS_PER_BLOCK + wave * ROWS_PER_WAVE;

  // ---- cache A fragments for 2 M-tiles: lane holds x[row][4s + 2h + {0,1}] ----
  v2f aF[2][16];
  #pragma unroll
  for (int m = 0; m < 2; ++m) {
    const float* xr = x + (rowBase + m * 16 + n) * LATENT_DIM + 2 * h;
    #pragma unroll
    for (int s = 0; s < 16; ++s)
      aF[m][s] = *(const v2f*)(xr + 4 * s);
  }

  float minv[2][8];
  int   mini[2][8];
  #pragma unroll
  for (int m = 0; m < 2; ++m)
    #pragma unroll
    for (int j = 0; j < 8; ++j) { minv[m][j] = 3.0e38f; mini[m][j] = 0; }

  // ---- sweep 64 code tiles: 16x16x64 via 16 chained K=4 f32 WMMAs ----
  for (int t = 0; t < 64; ++t) {
    v8f acc0 = {};
    v8f acc1 = {};
    const float* bbase = cbL + (t * 16 + n) * LATENT_DIM + 2 * h;
    #pragma unroll
    for (int s = 0; s < 16; ++s) {
      v2f b = *(const v2f*)(bbase + 4 * s);   // ds_load_b64, reused by both tiles
      acc0 = __builtin_amdgcn_wmma_f32_16x16x4_f32(
          false, aF[0][s], false, b, (short)0, acc0, false, false);
      acc1 = __builtin_amdgcn_wmma_f32_16x16x4_f32(
          false, aF[1][s], false, b, (short)0, acc1, false, false);
    }
    const float es  = esq[t * 16 + n];
    const int  code = t * 16 + n;
    #pragma unroll
    for (int j = 0; j < 8; ++j) {
      float s0 = fmaf(-2.0f, acc0[j], es);   // ||x||^2 constant per row: drop it
      if (s0 < minv[0][j]) { minv[0][j] = s0; mini[0][j] = code; }
      float s1 = fmaf(-2.0f, acc1[j], es);
      if (s1 < minv[1][j]) { minv[1][j] = s1; mini[1][j] = code; }
    }
  }

  // ---- argmin across the 16 lanes of each column group (wave32 butterfly) ----
  #pragma unroll
  for (int m = 0; m < 2; ++m) {
    #pragma unroll
    for (int j = 0; j < 8; ++j) {
      float v = minv[m][j];
      int idx = mini[m][j];
      #pragma unroll
      for (int off = 8; off >= 1; off >>= 1) {
        float ov = __shfl_xor(v, off, 32);
        int   oi = __shfl_xor(idx, off, 32);
        if (ov < v || (ov == v && oi < idx)) { v = ov; idx = oi; }  // jnp.argmin tie-break
      }
      if (n == 0)  // lanes 0 and 16 publish rows j+8h of M-tile m
        idxb[wave * ROWS_PER_WAVE + m * 16 + h * 8 + j] = idx;
    }
  }
  __syncthreads();

  // ---- outputs ----
  const long blockRow0 = (long)blockIdx.x * ROWS_PER_BLOCK;
  out_idx[blockRow0 + tid] = idxb[tid];            // coalesced int32 indices

  #pragma unroll
  for (int p = 0; p < 16; ++p) {                   // 16 threads x float4 per row
    const int r    = p * 16 + (tid >> 4);
    const int part = tid & 15;
    const int code = idxb[r];
    float4 v = *(const float4*)(cbL + code * LATENT_DIM + part * 4);
    *(float4*)(out_q + (blockRow0 + r) * LATENT_DIM + part * 4) = v;
  }
}

extern "C" void kernel_launch(void* const* d_in, const int* in_sizes, int n_in,
                              void* d_out, int out_size, void* d_ws, size_t ws_size,
                              hipStream_t stream) {
  const float* x  = (const float*)d_in[0];
  const float* cb = (const float*)d_in[1];
  const int N = in_sizes[0] / LATENT_DIM;          // 65536
  int*   out_idx = (int*)d_out;
  float* out_q   = (float*)d_out + N;

  const size_t shmem = LDS_FLOATS * sizeof(float); // 267264 bytes
  (void)hipFuncSetAttribute((const void*)vq_kernel,
                            hipFuncAttributeMaxDynamicSharedMemorySize,
                            (int)shmem);
  vq_kernel<<<dim3(N / ROWS_PER_BLOCK), dim3(BLOCK_THREADS), shmem, stream>>>(
      x, cb, out_idx, out_q);
}